// ScalableGNN_23227183137166
// MI455X (gfx1250) — compile-verified
//
#include <hip/hip_runtime.h>

// out = relu( relu( x[:B] @ W0 ) @ W1 ),  B = 80000, HIDDEN = 128, all fp32.
// CDNA5: V_WMMA_F32_16X16X4_F32 for the math, TENSOR_LOAD_TO_LDS (TDM) to stage W.

typedef float v2f __attribute__((ext_vector_type(2)));
typedef float v8f __attribute__((ext_vector_type(8)));
typedef unsigned int u32x4 __attribute__((ext_vector_type(4)));
typedef int i32x4 __attribute__((ext_vector_type(4)));
typedef int i32x8 __attribute__((ext_vector_type(8)));

#if defined(__HIP_DEVICE_COMPILE__) && !__has_builtin(__builtin_amdgcn_wmma_f32_16x16x4_f32)
#error "device pass: missing __builtin_amdgcn_wmma_f32_16x16x4_f32"
#endif

#if defined(__HIP_DEVICE_COMPILE__) && \
    __has_builtin(__builtin_amdgcn_tensor_load_to_lds) && \
    __has_builtin(__builtin_amdgcn_s_wait_tensorcnt)
#define HAVE_TDM 1
#else
#define HAVE_TDM 0
#endif

#define HIDDEN 128
#define ROWS_PER_BLOCK 128
#define WAVES 8
#define THREADS (WAVES * 32)
#define KCHUNKS (HIDDEN / 4)   // 32 K-steps of 4
#define NTILES (HIDDEN / 16)   // 8 column tiles

#if HAVE_TDM
// Build D# group 0: count=1 (valid), lds byte addr, 57-bit global byte addr, type=2.
__device__ inline u32x4 tdm_group0(unsigned lds_addr, unsigned long long gaddr) {
    u32x4 g;
    g.x = 1u;                                        // count=1, is_restore=0, gather=0
    g.y = lds_addr;                                  // [63:32] lds_addr
    g.z = (unsigned)(gaddr & 0xFFFFFFFFu);           // [95:64] global_addr lo
    g.w = ((unsigned)(gaddr >> 32) & 0x01FFFFFFu)    // [120:96] global_addr hi
          | (2u << 30);                              // [127:126] type = 2 ("image")
    return g;
}
// Build D# group 1: 2D tile, data_size=4B, no multicast/barrier/pad/iterate.
__device__ inline i32x8 tdm_group1(unsigned dim0, unsigned dim1,
                                   unsigned tile0, unsigned tile1,
                                   unsigned stride0) {
    i32x8 g;
    g[0] = (int)(2u << 16);                              // data_size=2 (4 bytes)
    g[1] = (int)((dim0 & 0xFFFFu) << 16);                // tensor_dim0[15:0] at bits[63:48]
    g[2] = (int)(((dim0 >> 16) & 0xFFFFu)                // tensor_dim0[31:16]
               | ((dim1 & 0xFFFFu) << 16));              // tensor_dim1[15:0]
    g[3] = (int)(((dim1 >> 16) & 0xFFFFu)                // tensor_dim1[31:16]
               | ((tile0 & 0xFFFFu) << 16));             // tile_dim0
    g[4] = (int)(tile1 & 0xFFFFu);                       // tile_dim1 (tile_dim2 = 0)
    g[5] = (int)stride0;                                 // tensor_dim0_stride[31:0]
    g[6] = 0;                                            // stride0 hi, stride1 lo
    g[7] = 0;                                            // stride1 hi
    return g;
}
#endif

__global__ __launch_bounds__(THREADS)
void gnn_fused_mlp2(const float* __restrict__ x,
                    const float* __restrict__ W0,
                    const float* __restrict__ W1,
                    float* __restrict__ out,
                    int nrows)
{
    extern __shared__ float smem[];
    float* sW0 = smem;                       // 128x128 f32 = 64 KB  @ lds 0
    float* sW1 = smem + HIDDEN * HIDDEN;     // 128x128 f32 = 64 KB  @ lds 65536
    float* sH  = smem + 2 * HIDDEN * HIDDEN; // 128x128 f32 = 64 KB  @ lds 131072

    const int tid  = threadIdx.x;
    const int wave = tid >> 5;
    const int lane = tid & 31;
    const int half = lane >> 4;   // 0: K-pair {0,1}, 1: K-pair {2,3}
    const int ln   = lane & 15;   // M index for A-frags, N index for B-frags

#if HAVE_TDM
    // ---- stage W0 / W1 into LDS via the Tensor Data Mover (async DMA) ----
    // TDM ignores EXEC and issues once per wave: let wave 0 own both descriptors.
    if (wave == 0) {
        u32x4 g0a = tdm_group0(0u, (unsigned long long)(uintptr_t)W0);
        u32x4 g0b = tdm_group0((unsigned)(HIDDEN * HIDDEN * sizeof(float)),
                               (unsigned long long)(uintptr_t)W1);
        i32x8 g1  = tdm_group1(HIDDEN, HIDDEN, HIDDEN, HIDDEN, HIDDEN);
        i32x4 gz4 = (i32x4)0;   // groups 2/3 unused for 2D tiles
        i32x8 gz8 = (i32x8)0;   // trailing group (clang-23 6-arg form)
        __builtin_amdgcn_tensor_load_to_lds(g0a, g1, gz4, gz4, gz8, 0);
        __builtin_amdgcn_tensor_load_to_lds(g0b, g1, gz4, gz4, gz8, 0);
        __builtin_amdgcn_s_wait_tensorcnt(0);
    }
#else
    {
        const float4* g0 = (const float4*)W0;
        const float4* g1 = (const float4*)W1;
        float4* l0 = (float4*)sW0;
        float4* l1 = (float4*)sW1;
        const int n4 = HIDDEN * HIDDEN / 4;
        for (int i = tid; i < n4; i += THREADS) {
            l0[i] = g0[i];
            l1[i] = g1[i];
        }
    }
#endif
    __syncthreads();

    const int rowBase = blockIdx.x * ROWS_PER_BLOCK + wave * 16;
    if (rowBase + 16 > nrows) return;  // whole-wave guard: EXEC stays all-ones

#if defined(__HIP_DEVICE_COMPILE__)
    // ---------------- Phase 1: H = relu(X @ W0) ----------------
    // A-frag (16x4 f32): lane ln holds row M=ln; VGPR0 = K=kb(+2*half), VGPR1 = K+1
    const float* xrow = x + (size_t)(rowBase + ln) * HIDDEN;

    v8f c[NTILES];
#pragma unroll
    for (int t = 0; t < NTILES; ++t) c[t] = (v8f)0.f;

    for (int kc = 0; kc < KCHUNKS; ++kc) {
        const int kb = kc * 4 + 2 * half;
        v2f a = *(const v2f*)(xrow + kb);          // 8B global load, reused x8
#pragma unroll
        for (int t = 0; t < NTILES; ++t) {
            v2f b;                                  // B-frag: K rows kb, kb+1 at N = t*16+ln
            b.x = sW0[(size_t)kb * HIDDEN + t * 16 + ln];
            b.y = sW0[(size_t)(kb + 1) * HIDDEN + t * 16 + ln];
            c[t] = __builtin_amdgcn_wmma_f32_16x16x4_f32(
                       false, a, false, b, (short)0, c[t], false, false);
        }
    }

    // ReLU + write wave-private 16x128 strip of H to LDS.
    // C layout: lanes 0-15 -> M = r (VGPR r), lanes 16-31 -> M = 8 + r; N = t*16 + ln.
    float* hbase = sH + (size_t)wave * 16 * HIDDEN;
#pragma unroll
    for (int t = 0; t < NTILES; ++t) {
#pragma unroll
        for (int r = 0; r < 8; ++r) {
            float v = c[t][r];
            v = v > 0.f ? v : 0.f;
            hbase[(half * 8 + r) * HIDDEN + t * 16 + ln] = v;
        }
    }
    // Rows are wave-private; same-wave LDS ops are in-order -> no barrier needed.

    // ---------------- Phase 2: OUT = relu(H @ W1) ----------------
    const float* hrow = sH + (size_t)(wave * 16 + ln) * HIDDEN;

    v8f d[NTILES];
#pragma unroll
    for (int t = 0; t < NTILES; ++t) d[t] = (v8f)0.f;

    for (int kc = 0; kc < KCHUNKS; ++kc) {
        const int kb = kc * 4 + 2 * half;
        v2f a = *(const v2f*)(hrow + kb);          // 8B LDS load
#pragma unroll
        for (int t = 0; t < NTILES; ++t) {
            v2f b;
            b.x = sW1[(size_t)kb * HIDDEN + t * 16 + ln];
            b.y = sW1[(size_t)(kb + 1) * HIDDEN + t * 16 + ln];
            d[t] = __builtin_amdgcn_wmma_f32_16x16x4_f32(
                       false, a, false, b, (short)0, d[t], false, false);
        }
    }

    // ReLU + store: each (t,r) writes 64 contiguous bytes per half-wave.
#pragma unroll
    for (int t = 0; t < NTILES; ++t) {
#pragma unroll
        for (int r = 0; r < 8; ++r) {
            float v = d[t][r];
            v = v > 0.f ? v : 0.f;
            out[(size_t)(rowBase + half * 8 + r) * HIDDEN + t * 16 + ln] = v;
        }
    }
#endif // __HIP_DEVICE_COMPILE__
}

extern "C" void kernel_launch(void* const* d_in, const int* in_sizes, int n_in,
                              void* d_out, int out_size, void* d_ws, size_t ws_size,
                              hipStream_t stream) {
    // setup_inputs order: 0:x 1:hist0 2:hist1 3:W0 4:W1 5:n_id 6:batch_size
    const float* x  = (const float*)d_in[0];
    const float* W0 = (const float*)d_in[3];
    const float* W1 = (const float*)d_in[4];
    float* out = (float*)d_out;

    const int nrows = out_size / HIDDEN;               // 80000
    const int blocks = (nrows + ROWS_PER_BLOCK - 1) / ROWS_PER_BLOCK;  // 625

    const size_t shmem =
        (size_t)(2 * HIDDEN * HIDDEN + ROWS_PER_BLOCK * HIDDEN) * sizeof(float); // 192 KB

    (void)hipFuncSetAttribute(reinterpret_cast<const void*>(gnn_fused_mlp2),
                              hipFuncAttributeMaxDynamicSharedMemorySize,
                              (int)shmem);

    gnn_fused_mlp2<<<blocks, THREADS, shmem, stream>>>(x, W0, W1, out, nrows);
}